// DynamicAttention_54597624267060
// MI455X (gfx1250) — compile-verified
//
#include <hip/hip_runtime.h>
#include <hip/hip_bf16.h>
#include <math.h>

#define NUM_GRAPHS 1024
#define IN_DIM 256
#define HID 128
#define N_NODES 200000

typedef float v2f __attribute__((ext_vector_type(2)));
typedef float v8f __attribute__((ext_vector_type(8)));

// ---------------------------------------------------------------------------
// Kernel 1: segment-mean pool. batch is sorted int64, so each graph's nodes
// are a contiguous range -> binary search boundaries, no atomics.
// One block per graph, one thread per feature dim (coalesced row reads).
// ---------------------------------------------------------------------------
__global__ void dynattn_pool_kernel(const float* __restrict__ x,
                                    const long long* __restrict__ batch,
                                    float* __restrict__ graph_repr) {
    const int g = blockIdx.x;
    const int d = threadIdx.x;  // 256 threads == IN_DIM

    // lower_bound: first index i with batch[i] >= v  (redundant per-thread,
    // but all threads hit identical addresses -> scalar-cached broadcast)
    int lo = 0, hi = N_NODES;
    while (lo < hi) { int mid = (lo + hi) >> 1; if (batch[mid] < (long long)g) lo = mid + 1; else hi = mid; }
    const int start = lo;
    lo = start; hi = N_NODES;
    while (lo < hi) { int mid = (lo + hi) >> 1; if (batch[mid] < (long long)(g + 1)) lo = mid + 1; else hi = mid; }
    const int end = lo;

    float acc = 0.0f;
    for (int n = start; n < end; ++n) {
        acc += x[(size_t)n * IN_DIM + d];
    }
    const int cnt = end - start;
    const float inv = 1.0f / (float)(cnt > 0 ? cnt : 1);
    graph_repr[(size_t)g * IN_DIM + d] = acc * inv;
}

// ---------------------------------------------------------------------------
// Kernel 2: H = PReLU(graph_repr @ W1 + b1) via V_WMMA_F32_16X16X4_F32.
// M=1024, K=256, N=128. Grid = 64 blocks of 8 waves (256 threads);
// wave w handles output tile (mTile = blockIdx.x, nTile = w).
// A fragment (16x4 f32, 2 VGPRs/lane): lane L holds row m=L%16,
//   K = (L/16)*2 + {0,1}. B fragment (4x16) is the mirror: lane L holds
//   col n=L%16, K = (L/16)*2 + {0,1}.
// C/D (16x16 f32, 8 VGPRs): vgpr j, lane L -> m = j + 8*(L/16), n = L%16.
// ---------------------------------------------------------------------------
__global__ void dynattn_mlp1_wmma_kernel(const float* __restrict__ A,    // [1024,256]
                                         const float* __restrict__ W1,   // [256,128] (in,out)
                                         const float* __restrict__ b1,   // [128]
                                         const float* __restrict__ prelu_a,
                                         float* __restrict__ H) {        // [1024,128]
    const int mTile = blockIdx.x;          // 0..63
    const int nTile = threadIdx.x >> 5;    // wave id 0..7
    const int lane  = threadIdx.x & 31;
    const int lmod  = lane & 15;
    const int lhi   = lane >> 4;           // 0 or 1

    const int m0 = mTile * 16;
    const int n0 = nTile * 16;

    v8f c = {};
    for (int k = 0; k < IN_DIM; k += 4) {
        const int ka = k + lhi * 2;
        v2f a, b;
        const float* arow = A + (size_t)(m0 + lmod) * IN_DIM + ka;
        a.x = arow[0];
        a.y = arow[1];
        const float* bcol = W1 + (size_t)ka * HID + n0 + lmod;
        b.x = bcol[0];
        b.y = bcol[HID];
        // (neg_a, A, neg_b, B, c_mod, C, reuse_a, reuse_b)
        c = __builtin_amdgcn_wmma_f32_16x16x4_f32(false, a, false, b,
                                                  (short)0, c, false, false);
    }

    const float alpha = prelu_a[0];
    const int n = n0 + lmod;
    const float bias = b1[n];
#pragma unroll
    for (int j = 0; j < 8; ++j) {
        const int m = m0 + j + 8 * lhi;
        float h = c[j] + bias;
        h = (h >= 0.0f) ? h : alpha * h;
        H[(size_t)m * HID + n] = h;
    }
}

// ---------------------------------------------------------------------------
// Kernel 3: attn[g] = sigmoid(dot(H[g,:], W2) + b2). One block per graph.
// ---------------------------------------------------------------------------
__global__ void dynattn_attn_kernel(const float* __restrict__ H,
                                    const float* __restrict__ W2,
                                    const float* __restrict__ b2,
                                    float* __restrict__ attn) {
    const int g = blockIdx.x;
    const int t = threadIdx.x;  // 128
    __shared__ float red[HID];
    red[t] = H[(size_t)g * HID + t] * W2[t];
    __syncthreads();
#pragma unroll
    for (int s = HID / 2; s > 0; s >>= 1) {
        if (t < s) red[t] += red[t + s];
        __syncthreads();
    }
    if (t == 0) {
        const float z = red[0] + b2[0];
        attn[g] = 1.0f / (1.0f + expf(-z));
    }
}

// ---------------------------------------------------------------------------
// Kernel 4: out = x * attn[batch], float4-vectorized. 4 nodes per 256-thread
// block (64 float4 lanes per node row).
// ---------------------------------------------------------------------------
__global__ void dynattn_scale_kernel(const float* __restrict__ x,
                                     const long long* __restrict__ batch,
                                     const float* __restrict__ attn,
                                     float* __restrict__ out) {
    const int node = blockIdx.x * 4 + (threadIdx.x >> 6);
    if (node >= N_NODES) return;
    const int q = threadIdx.x & 63;
    const float s = attn[(int)batch[node]];
    const float4* xin = (const float4*)(x + (size_t)node * IN_DIM);
    float4* o = (float4*)(out + (size_t)node * IN_DIM);
    float4 v = xin[q];
    v.x *= s; v.y *= s; v.z *= s; v.w *= s;
    o[q] = v;
}

// ---------------------------------------------------------------------------
extern "C" void kernel_launch(void* const* d_in, const int* in_sizes, int n_in,
                              void* d_out, int out_size, void* d_ws, size_t ws_size,
                              hipStream_t stream) {
    const float*     x       = (const float*)d_in[0];
    const long long* batch   = (const long long*)d_in[1];
    const float*     W1      = (const float*)d_in[2];
    const float*     b1      = (const float*)d_in[3];
    const float*     prelu_a = (const float*)d_in[4];
    const float*     W2      = (const float*)d_in[5];
    const float*     b2      = (const float*)d_in[6];
    float*           out     = (float*)d_out;

    float* graph_repr = (float*)d_ws;                       // 1024*256 f32
    float* H          = graph_repr + NUM_GRAPHS * IN_DIM;   // 1024*128 f32
    float* attn       = H + NUM_GRAPHS * HID;               // 1024 f32

    dynattn_pool_kernel<<<NUM_GRAPHS, IN_DIM, 0, stream>>>(x, batch, graph_repr);
    dynattn_mlp1_wmma_kernel<<<NUM_GRAPHS / 16, 256, 0, stream>>>(graph_repr, W1, b1, prelu_a, H);
    dynattn_attn_kernel<<<NUM_GRAPHS, HID, 0, stream>>>(H, W2, b2, attn);
    dynattn_scale_kernel<<<(N_NODES + 3) / 4, 256, 0, stream>>>(x, batch, attn, out);
}